// CosineProxy_29351806500968
// MI455X (gfx1250) — compile-verified
//
#include <hip/hip_runtime.h>
#include <hip/hip_bf16.h>

// Problem: B=256, N=5, C=640, H=10, W=10 ; pool -> (32,5,5)=800 dims
#define BATCH 256
#define NSHOT 5
#define SHOT  64000            // C*H*W
#define XB    (NSHOT * SHOT)   // elements of x per batch = 320000
#define DPOOL 800              // 32*5*5
#define SHOT4 (SHOT / 4)       // 16000 float4 per shot

// feats padded to 16 rows (WMMA M=16) and stride 802 floats:
//  - rows 5..15 zeroed -> unconditional ds_load_b64 in the Gram K-loop (no exec masking)
//  - 802 mod 64 = 34 -> all 16 row bases hit distinct LDS banks (vs 8-way conflict at 800)
#define MROWS   16
#define DSTRIDE 802

typedef __attribute__((ext_vector_type(2))) float v2f;
typedef __attribute__((ext_vector_type(8))) float v8f;

// One workgroup per batch element b.
//   Phase A: pool x[b] (1.28MB, coalesced float2) -> feats[5][800] in LDS (unscaled by 1/80;
//            cosine similarity is scale invariant, only eps rescales: 1e-8 * 80^2 = 6.4e-5)
//   Phase B: wave 0 computes the 5x5 Gram G = F * F^T with V_WMMA_F32_16X16X4_F32
//            (K=800, 200 steps). For 32-bit data the ISA A(16x4) and B(4x16) lane layouts
//            coincide (lane%16 = row/col index, K split over {vgpr, lane>=16}), so one
//            v2f register pair feeds both operands.
//            dot[n] = sum_m G[m][n], na[n] = sqrt(G[n][n]), nb = sqrt(sum G).
//   Phase C: out[b] = sum_n sim[n] * x[b,n]  (float4 streaming; x[b] re-read hits L2).
__global__ __launch_bounds__(256) void CosineProxy_kernel(const float* __restrict__ x,
                                                          float* __restrict__ out) {
    const int b   = blockIdx.x;
    const int tid = threadIdx.x;

    __shared__ float feats[MROWS][DSTRIDE];   // ~51 KB
    __shared__ float s_dot[8];
    __shared__ float s_diag[8];
    __shared__ float s_sim[8];

    const float* __restrict__ xb = x + (size_t)b * XB;

    // Zero the padding rows 5..15 (k < DPOOL is all the Gram loop reads).
    for (int i = tid; i < (MROWS - NSHOT) * DPOOL; i += 256) {
        feats[NSHOT + i / DPOOL][i % DPOOL] = 0.f;
    }

    // ---------------- Phase A: adaptive avg pool (unscaled sums) ----------------
    // d = (oc, oh, ow); each pooled value sums a 20(ch) x 2(h) x 2(w) block.
    for (int d = tid; d < DPOOL; d += 256) {
        const int oc = d / 25;
        const int r  = d % 25;
        const int oh = r / 5;
        const int ow = r % 5;
        const int base = oc * 2000 + oh * 20 + ow * 2;   // always even -> float2 aligned
#pragma unroll
        for (int n = 0; n < NSHOT; ++n) {
            const float2* __restrict__ p2 =
                reinterpret_cast<const float2*>(xb + n * SHOT + base);
            float acc = 0.f;
#pragma unroll
            for (int i = 0; i < 20; ++i) {               // channel within group
                float2 u = p2[i * 50];                   // h = 2*oh,   w = 2*ow..2*ow+1
                float2 v = p2[i * 50 + 5];               // h = 2*oh+1
                acc += (u.x + u.y) + (v.x + v.y);
            }
            feats[n][d] = acc;
        }
    }
    __syncthreads();

    // ---------------- Phase B: 5x5 Gram via WMMA f32 16x16x4 (wave 0) ----------------
    if (tid < 32) {
        const int lane = tid;
        const int m16  = lane & 15;        // row index of A / col index of B
        const int kh   = (lane >> 4) << 1; // lanes 0-15 -> K = k0,k0+1 ; lanes 16-31 -> k0+2,k0+3
        const float* __restrict__ frow = &feats[m16][kh];

        v8f c = {};
        for (int k0 = 0; k0 < DPOOL; k0 += 4) {
            // unconditional 8-byte LDS load; rows >= NSHOT are zero padding
            const v2f a = *reinterpret_cast<const v2f*>(frow + k0);
            // D = A * B + C ; A == B register image -> D = F * F^T (Gram), fp32 throughout
            c = __builtin_amdgcn_wmma_f32_16x16x4_f32(
                    /*neg_a=*/false, a, /*neg_b=*/false, a,
                    /*c_mod=*/(short)0, c, /*reuse_a=*/false, /*reuse_b=*/false);
        }
        // C/D layout: VGPR m, lanes 0-15 -> G[m][n=lane]; rows m=0..4 are the valid ones.
        if (lane < NSHOT) {
            const float dot  = c[0] + c[1] + c[2] + c[3] + c[4];
            const float diag = (lane == 0) ? c[0]
                             : (lane == 1) ? c[1]
                             : (lane == 2) ? c[2]
                             : (lane == 3) ? c[3] : c[4];
            s_dot[lane]  = dot;
            s_diag[lane] = diag;
        }
    }
    __syncthreads();

    if (tid < NSHOT) {
        const float nb2 = s_dot[0] + s_dot[1] + s_dot[2] + s_dot[3] + s_dot[4];
        const float nb  = sqrtf(nb2);                    // ||proxy|| (unscaled)
        const float na  = sqrtf(s_diag[tid]);            // ||feats[n]|| (unscaled)
        // eps rescaled: reference max(na*nb, 1e-8) on /80-scaled feats -> 1e-8 * 6400
        s_sim[tid] = s_dot[tid] / fmaxf(na * nb, 6.4e-5f);
    }
    __syncthreads();

    // ---------------- Phase C: out[b] = sum_n sim[n] * x[b,n] ----------------
    const float s0 = s_sim[0], s1 = s_sim[1], s2 = s_sim[2], s3 = s_sim[3], s4 = s_sim[4];
    const float4* __restrict__ x4 = reinterpret_cast<const float4*>(xb);
    float4* __restrict__ o4 = reinterpret_cast<float4*>(out + (size_t)b * SHOT);

    for (int i = tid; i < SHOT4; i += 256) {
        float4 a0 = x4[i];
        float4 a1 = x4[i + 1 * SHOT4];
        float4 a2 = x4[i + 2 * SHOT4];
        float4 a3 = x4[i + 3 * SHOT4];
        float4 a4 = x4[i + 4 * SHOT4];
        float4 o;
        o.x = s0 * a0.x + s1 * a1.x + s2 * a2.x + s3 * a3.x + s4 * a4.x;
        o.y = s0 * a0.y + s1 * a1.y + s2 * a2.y + s3 * a3.y + s4 * a4.y;
        o.z = s0 * a0.z + s1 * a1.z + s2 * a2.z + s3 * a3.z + s4 * a4.z;
        o.w = s0 * a0.w + s1 * a1.w + s2 * a2.w + s3 * a3.w + s4 * a4.w;
        o4[i] = o;
    }
}

extern "C" void kernel_launch(void* const* d_in, const int* in_sizes, int n_in,
                              void* d_out, int out_size, void* d_ws, size_t ws_size,
                              hipStream_t stream) {
    const float* x = (const float*)d_in[0];   // [256, 5, 640, 10, 10] fp32
    float* out = (float*)d_out;               // [256, 640, 10, 10] fp32
    (void)in_sizes; (void)n_in; (void)out_size; (void)d_ws; (void)ws_size;
    CosineProxy_kernel<<<dim3(BATCH), dim3(256), 0, stream>>>(x, out);
}